// GraphAttentionEmbedding_89850715832321
// MI455X (gfx1250) — compile-verified
//
#include <hip/hip_runtime.h>
#include <hip/hip_bf16.h>
#include <math.h>

#define NN 50000
#define EE 640000
#define DD 128
#define EDIM 256

typedef float v2f __attribute__((ext_vector_type(2)));
typedef float v8f __attribute__((ext_vector_type(8)));

// ---------------------------------------------------------------------------
// Kernel A: node projections  qn,kn,vn -> ws ; sn -> d_out (skip connection)
// One 16-node tile per block; 8 waves: wave -> (matrix = w&3, col-half = w>>2)
// WMMA f32 16x16x4, K=128 -> 32 steps per 16x16 tile, 4 tiles per wave.
// ---------------------------------------------------------------------------
__global__ __launch_bounds__(256)
void node_proj_kernel(const float* __restrict__ x,
                      const float* __restrict__ Wq, const float* __restrict__ bq,
                      const float* __restrict__ Wk, const float* __restrict__ bk,
                      const float* __restrict__ Wv, const float* __restrict__ bv,
                      const float* __restrict__ Ws, const float* __restrict__ bs,
                      float* __restrict__ qn, float* __restrict__ kn,
                      float* __restrict__ vn, float* __restrict__ out)
{
    const int tile  = blockIdx.x;           // 16-node tile, 3125 total
    const int wave  = threadIdx.x >> 5;     // 0..7
    const int lane  = threadIdx.x & 31;
    const int mat   = wave & 3;
    const int cth   = wave >> 2;            // 0: ct 0..3, 1: ct 4..7

    const float* W; const float* bias; float* dst;
    if      (mat == 0) { W = Wq; bias = bq; dst = qn;  }
    else if (mat == 1) { W = Wk; bias = bk; dst = kn;  }
    else if (mat == 2) { W = Wv; bias = bv; dst = vn;  }
    else               { W = Ws; bias = bs; dst = out; }

    const int row   = lane & 15;
    const int khalf = lane >> 4;            // 0 or 1
    const int node0 = tile * 16;

    // A fragments: x[16 rows][K=128]; frag kk: lane -> cols 4kk+2*khalf .. +1
    v2f a[32];
    const float* xrow = x + (size_t)(node0 + row) * DD;
    #pragma unroll
    for (int kk = 0; kk < 32; ++kk) {
        const float* p = xrow + kk * 4 + khalf * 2;
        a[kk] = v2f{p[0], p[1]};
    }

    #pragma unroll
    for (int c = 0; c < 4; ++c) {
        const int ct   = cth * 4 + c;
        const int ncol = ct * 16 + row;
        v8f acc = {};
        #pragma unroll
        for (int kk = 0; kk < 32; ++kk) {
            const int kb = kk * 4 + khalf * 2;
            v2f b = v2f{W[kb * DD + ncol], W[(kb + 1) * DD + ncol]};
            acc = __builtin_amdgcn_wmma_f32_16x16x4_f32(
                false, a[kk], false, b, (short)0, acc, false, false);
        }
        const float bb = bias[ncol];
        #pragma unroll
        for (int j = 0; j < 8; ++j) {
            const int m = j + khalf * 8;
            dst[(size_t)(node0 + m) * DD + ncol] = acc[j] + bb;
        }
    }
}

// ---------------------------------------------------------------------------
// Init: segment-max keys to -inf (INT_MIN in order-preserving int space),
// denominators to zero.
// ---------------------------------------------------------------------------
__global__ __launch_bounds__(256)
void init_kernel(int* __restrict__ mkey, float* __restrict__ denom, int n)
{
    int i = blockIdx.x * 256 + threadIdx.x;
    if (i < n) { mkey[i] = (int)0x80000000; denom[i] = 0.0f; }
}

// ---------------------------------------------------------------------------
// Kernel B1: e = [cos(rel_t*te_w+te_b) | msg] @ We   -> ebuf[E,128]
// 2 waves/block, wave per 16-edge tile. edge_attr tile staged in LDS
// (stride 264 floats for bank spread). Phase 2 is kk-outer / ct-inner:
// each rolled iteration issues 8 INDEPENDENT WMMAs (8 accumulator chains),
// reads the A fragment from LDS once, and clauses the 16 We loads.
// ---------------------------------------------------------------------------
__global__ __launch_bounds__(64)
void edge_gemm_kernel(const int* __restrict__ srcIdx,
                      const float* __restrict__ lastUpd,
                      const float* __restrict__ tEdge,
                      const float* __restrict__ msg,
                      const float* __restrict__ te_w,
                      const float* __restrict__ te_b,
                      const float* __restrict__ We,
                      float* __restrict__ ebuf)
{
    __shared__ float smem[2][16 * 264];
    const int wave  = threadIdx.x >> 5;     // 0..1
    const int lane  = threadIdx.x & 31;
    const int tile  = blockIdx.x * 2 + wave;
    const int ebase = tile * 16;
    float* sm = &smem[wave][0];

    // ---- Phase 1: fill edge_attr tile (16 x 256) in LDS ----
    {
        const int r    = lane >> 1;         // row 0..15
        const int half = lane & 1;          // 0: time-encoding cols, 1: msg cols
        const int e    = ebase + r;
        if (half == 0) {
            const int   s   = srcIdx[e];
            const float rel = lastUpd[s] - tEdge[e];
            for (int j = 0; j < 128; j += 4) {
                float4 w4 = *(const float4*)(te_w + j);
                float4 b4 = *(const float4*)(te_b + j);
                float4 te4;
                te4.x = cosf(rel * w4.x + b4.x);
                te4.y = cosf(rel * w4.y + b4.y);
                te4.z = cosf(rel * w4.z + b4.z);
                te4.w = cosf(rel * w4.w + b4.w);
                *(float4*)&sm[r * 264 + j] = te4;
            }
        } else {
            const float4* mrow = (const float4*)(msg + (size_t)e * 128);
            for (int j = 0; j < 128; j += 4)
                *(float4*)&sm[r * 264 + 128 + j] = mrow[j >> 2];
        }
    }
    __syncthreads();

    // ---- Phase 2: WMMA GEMM, A from LDS, B = We (L2-resident 128 KB) ----
    const int r16   = lane & 15;
    const int khalf = lane >> 4;

    v8f acc[8];
    #pragma unroll
    for (int ct = 0; ct < 8; ++ct) acc[ct] = v8f{};

    for (int kk = 0; kk < 64; ++kk) {
        v2f a = *(const v2f*)&sm[r16 * 264 + kk * 4 + khalf * 2];
        const int kb = kk * 4 + khalf * 2;
        const float* Wrow0 = We + (size_t)kb * DD;        // row kb
        const float* Wrow1 = Wrow0 + DD;                  // row kb+1
        #pragma unroll
        for (int ct = 0; ct < 8; ++ct) {
            const int ncol = ct * 16 + r16;
            v2f b = v2f{Wrow0[ncol], Wrow1[ncol]};
            acc[ct] = __builtin_amdgcn_wmma_f32_16x16x4_f32(
                false, a, false, b, (short)0, acc[ct], false, false);
        }
    }

    #pragma unroll
    for (int ct = 0; ct < 8; ++ct) {
        const int ncol = ct * 16 + r16;
        #pragma unroll
        for (int j = 0; j < 8; ++j)
            ebuf[(size_t)(ebase + j + khalf * 8) * DD + ncol] = acc[ct][j];
    }
}

// ---------------------------------------------------------------------------
// Kernel B2: wave per edge. k = kn[src]+e, v = vn[src]+e (overwrites ebuf),
// logits = <q[dst],k>/8 per head (half-wave shuffle reduce), atomicMax m[dst].
// ---------------------------------------------------------------------------
__global__ __launch_bounds__(256)
void edge_qkv_kernel(const int* __restrict__ srcIdx, const int* __restrict__ dstIdx,
                     const float* __restrict__ qn, const float* __restrict__ kn,
                     const float* __restrict__ vn,
                     float* __restrict__ ebuf, float* __restrict__ lgt,
                     int* __restrict__ mkey)
{
    const int e    = blockIdx.x * 8 + (threadIdx.x >> 5);
    const int lane = threadIdx.x & 31;
    const int s    = srcIdx[e];
    const int d    = dstIdx[e];
    const int c0   = lane * 4;              // 4 channels per lane, head = lane>>4

    float4 ev = *(const float4*)(ebuf + (size_t)e * DD + c0);
    float4 kv = *(const float4*)(kn   + (size_t)s * DD + c0);
    float4 vv = *(const float4*)(vn   + (size_t)s * DD + c0);
    float4 qv = *(const float4*)(qn   + (size_t)d * DD + c0);

    float4 k4 = {kv.x + ev.x, kv.y + ev.y, kv.z + ev.z, kv.w + ev.w};
    float4 v4 = {vv.x + ev.x, vv.y + ev.y, vv.z + ev.z, vv.w + ev.w};
    *(float4*)(ebuf + (size_t)e * DD + c0) = v4;   // v_e replaces e in place

    float dot = qv.x * k4.x + qv.y * k4.y + qv.z * k4.z + qv.w * k4.w;
    #pragma unroll
    for (int off = 1; off < 16; off <<= 1)
        dot += __shfl_xor(dot, off, 32);    // reduce within each 16-lane head

    if (lane == 0 || lane == 16) {
        const int   h     = lane >> 4;
        const float logit = dot * 0.125f;   // 1/sqrt(64)
        lgt[e * 2 + h] = logit;
        int b   = __float_as_int(logit);
        int key = b ^ ((b >> 31) & 0x7fffffff);   // order-preserving int map
        atomicMax(&mkey[d * 2 + h], key);
    }
}

// ---------------------------------------------------------------------------
// Kernel C: ex = exp(logit - m[dst]) in place; denom[dst] += ex
// ---------------------------------------------------------------------------
__global__ __launch_bounds__(256)
void softmax_num_kernel(const int* __restrict__ dstIdx,
                        float* __restrict__ lgt,
                        const int* __restrict__ mkey,
                        float* __restrict__ denom)
{
    const int i = blockIdx.x * 256 + threadIdx.x;  // over E*2
    if (i >= EE * 2) return;
    const int e = i >> 1, h = i & 1;
    const int d = dstIdx[e];
    const int key = mkey[d * 2 + h];
    const float mmax = __int_as_float(key ^ ((key >> 31) & 0x7fffffff));
    const float ex = expf(lgt[i] - mmax);
    lgt[i] = ex;
    atomicAdd(&denom[d * 2 + h], ex);
}

// ---------------------------------------------------------------------------
// Kernel D: wave per edge; out[dst] += v_e * (ex / denom[dst])
// ---------------------------------------------------------------------------
__global__ __launch_bounds__(256)
void scatter_kernel(const int* __restrict__ dstIdx,
                    const float* __restrict__ ve,
                    const float* __restrict__ exbuf,
                    const float* __restrict__ denom,
                    float* __restrict__ out)
{
    const int e    = blockIdx.x * 8 + (threadIdx.x >> 5);
    const int lane = threadIdx.x & 31;
    const int d    = dstIdx[e];
    const int h    = lane >> 4;
    const float alpha = exbuf[e * 2 + h] / denom[d * 2 + h];
    const int c0 = lane * 4;
    float4 v4 = *(const float4*)(ve + (size_t)e * DD + c0);
    float* o = out + (size_t)d * DD + c0;
    atomicAdd(o + 0, v4.x * alpha);
    atomicAdd(o + 1, v4.y * alpha);
    atomicAdd(o + 2, v4.z * alpha);
    atomicAdd(o + 3, v4.w * alpha);
}

// ---------------------------------------------------------------------------
extern "C" void kernel_launch(void* const* d_in, const int* in_sizes, int n_in,
                              void* d_out, int out_size, void* d_ws, size_t ws_size,
                              hipStream_t stream)
{
    const float* x    = (const float*)d_in[0];
    const float* lu   = (const float*)d_in[1];
    const int*   ei   = (const int*)  d_in[2];   // (2,E): src then dst
    const float* t    = (const float*)d_in[3];
    const float* msg  = (const float*)d_in[4];
    const float* te_w = (const float*)d_in[5];
    const float* te_b = (const float*)d_in[6];
    const float* Wq   = (const float*)d_in[7];
    const float* bq   = (const float*)d_in[8];
    const float* Wk   = (const float*)d_in[9];
    const float* bk   = (const float*)d_in[10];
    const float* Wv   = (const float*)d_in[11];
    const float* bv   = (const float*)d_in[12];
    const float* We   = (const float*)d_in[13];
    const float* Ws_  = (const float*)d_in[14];
    const float* bs   = (const float*)d_in[15];

    const int* srcIdx = ei;
    const int* dstIdx = ei + EE;

    float* out = (float*)d_out;

    // workspace layout (floats)
    float* qn    = (float*)d_ws;
    float* kn    = qn + (size_t)NN * DD;
    float* vn    = kn + (size_t)NN * DD;
    float* ebuf  = vn + (size_t)NN * DD;               // e, later v_e  [E,128]
    float* lgt   = ebuf + (size_t)EE * DD;             // logits, later ex [E,2]
    int*   mkey  = (int*)(lgt + (size_t)EE * 2);       // [N,2]
    float* denom = (float*)(mkey + (size_t)NN * 2);    // [N,2]

    // A: node projections (writes skip projection into d_out)
    node_proj_kernel<<<dim3(NN / 16), dim3(256), 0, stream>>>(
        x, Wq, bq, Wk, bk, Wv, bv, Ws_, bs, qn, kn, vn, out);

    // init segment max / denom
    init_kernel<<<dim3((NN * 2 + 255) / 256), dim3(256), 0, stream>>>(
        mkey, denom, NN * 2);

    // B1: edge-attr GEMM (WMMA) -> ebuf
    edge_gemm_kernel<<<dim3(EE / 32), dim3(64), 0, stream>>>(
        srcIdx, lu, t, msg, te_w, te_b, We, ebuf);

    // B2: k/v/logits + segment max
    edge_qkv_kernel<<<dim3(EE / 8), dim3(256), 0, stream>>>(
        srcIdx, dstIdx, qn, kn, vn, ebuf, lgt, mkey);

    // C: exp + denominator
    softmax_num_kernel<<<dim3((EE * 2 + 255) / 256), dim3(256), 0, stream>>>(
        dstIdx, lgt, mkey, denom);

    // D: weighted scatter into d_out
    scatter_kernel<<<dim3(EE / 8), dim3(256), 0, stream>>>(
        dstIdx, ebuf, lgt, denom, out);
}